// ForwardDeformer_58858231824614
// MI455X (gfx1250) — compile-verified
//
#include <hip/hip_runtime.h>
#include <math.h>

#define DD   32
#define HH   128
#define WW   128
#define DHW  (DD*HH*WW)
#define NJ   24
#define NI   13
#define NITER 10
#define RATIO 4.0f
#define CVG  1e-5f
#define DVG  0.1f
#define DUP_EPS2 1e-8f

typedef __attribute__((ext_vector_type(2))) float v2f;
typedef __attribute__((ext_vector_type(8))) float v8f;

__device__ __constant__ int c_init_bones[NI] = {0,1,2,4,5,10,11,12,15,16,17,18,19};

__device__ inline void inv3x3(const float R[9], float o[9]) {
    float a=R[0],b=R[1],c=R[2],d=R[3],e=R[4],f=R[5],g=R[6],h=R[7],i=R[8];
    float A =  (e*i - f*h);
    float B = -(d*i - f*g);
    float C =  (d*h - e*g);
    float det = a*A + b*B + c*C;
    float id = 1.0f / det;
    o[0]=A*id;           o[1]=-(b*i - c*h)*id;  o[2]= (b*f - c*e)*id;
    o[3]=B*id;           o[4]= (a*i - c*g)*id;  o[5]=-(a*f - c*d)*id;
    o[6]=C*id;           o[7]=-(a*h - b*g)*id;  o[8]= (a*e - b*d)*id;
}

// ---------------- tinv: invert tfs[INIT_BONES] (4x4 with [0,0,0,1] bottom row) ----------------
__global__ void k_tinv(const float* __restrict__ tfs, float* __restrict__ out) {
    int i = threadIdx.x;
    if (i >= NI) return;
    int b = c_init_bones[i];
    const float* T = tfs + b * 16;
    float R[9] = {T[0],T[1],T[2], T[4],T[5],T[6], T[8],T[9],T[10]};
    float t0=T[3], t1=T[7], t2=T[11];
    float inv[9]; inv3x3(R, inv);
    float v0 = -(inv[0]*t0 + inv[1]*t1 + inv[2]*t2);
    float v1 = -(inv[3]*t0 + inv[4]*t1 + inv[5]*t2);
    float v2 = -(inv[6]*t0 + inv[7]*t1 + inv[8]*t2);
    float* o = out + i * 12;
    o[0]=inv[0]; o[1]=inv[1]; o[2]=inv[2];  o[3]=v0;
    o[4]=inv[3]; o[5]=inv[4]; o[6]=inv[5];  o[7]=v1;
    o[8]=inv[6]; o[9]=inv[7]; o[10]=inv[8]; o[11]=v2;
}

// ---------------- precompute voxel_d (3ch) and voxel_J (9ch) tables ----------------
// T(v, n) = sum_j lbs[j, v] * tfs[j][n]  for n in 0..11  --> fp32 WMMA 16x16x4, K=24 in 6 steps.
// Each wave computes 16 voxel rows; result transposed through LDS so one lane owns one voxel.
// B columns 12..15 deliberately carry the tfs bottom row [0,0,0,1] (in-bounds reads): those
// output columns are never consumed, so no predication/select is needed -> branch-free loads.
__global__ void __launch_bounds__(256) k_precompute(
    const float* __restrict__ lbs, const float* __restrict__ tfs,
    const float* __restrict__ offset, const float* __restrict__ scale,
    float* __restrict__ vd_out, float* __restrict__ vj_out)
{
    __shared__ float lds[8][16][16];
    const int lane = threadIdx.x & 31;
    const int wv   = threadIdx.x >> 5;
    const int vb   = (blockIdx.x * 8 + wv) * 16;   // 16 voxels per wave
    const int mrow = lane & 15;
    const int kk   = (lane >> 4) << 1;             // 0 for lanes 0-15, 2 for lanes 16-31

    v8f acc = {};
    #pragma unroll
    for (int kc = 0; kc < 6; ++kc) {
        const int k0 = kc * 4 + kk;
        v2f a, b;
        // A: 16x4 fp32 tile, lanes 0-15 hold K=k0,k0+1; lanes 16-31 hold K=k0+2,k0+3
        a[0] = lbs[(size_t)(k0    ) * DHW + vb + mrow];
        a[1] = lbs[(size_t)(k0 + 1) * DHW + vb + mrow];
        // B: 4x16 fp32 tile, column n = flattened entry n of tfs[j] (row-major 4x4)
        b[0] = tfs[(k0    ) * 16 + mrow];
        b[1] = tfs[(k0 + 1) * 16 + mrow];
        acc = __builtin_amdgcn_wmma_f32_16x16x4_f32(false, a, false, b, (short)0, acc, false, false);
    }

    // D layout: VGPR r -> rows (r, r+8) for lane groups (0-15, 16-31), col = lane&15
    const int mhi = (lane >> 4) << 3;
    #pragma unroll
    for (int r = 0; r < 8; ++r)
        lds[wv][r + mhi][mrow] = acc[r];
    __syncthreads();

    const int t = threadIdx.x;
    if (t < 128) {
        const int v = blockIdx.x * 128 + t;
        float T[12];
        #pragma unroll
        for (int n = 0; n < 12; ++n) T[n] = lds[t >> 4][t & 15][n];

        const int xi = v & (WW - 1);
        const int yi = (v >> 7) & (HH - 1);
        const int zi = v >> 14;
        const float sc = scale[0];
        const float px = (-1.0f + 2.0f * xi * (1.0f / (WW - 1))) * sc + offset[0];
        const float py = (-1.0f + 2.0f * yi * (1.0f / (HH - 1))) * sc + offset[1];
        const float pz = (-1.0f + 2.0f * zi * (1.0f / (DD - 1))) * (1.0f / RATIO) * sc + offset[2];

        #pragma unroll
        for (int a = 0; a < 3; ++a) {
            float vd = T[4*a+0]*px + T[4*a+1]*py + T[4*a+2]*pz + T[4*a+3];
            vd_out[(size_t)a * DHW + v] = vd;
        }
        float R[9] = {T[0],T[1],T[2], T[4],T[5],T[6], T[8],T[9],T[10]};
        float inv[9]; inv3x3(R, inv);
        #pragma unroll
        for (int c = 0; c < 9; ++c)
            vj_out[(size_t)c * DHW + v] = inv[c];
    }
}

// ---------------- trilinear helpers ----------------
struct Tri { int idx[8]; float w[8]; };

__device__ inline Tri tri_make(float px, float py, float pz,
                               float o0, float o1, float o2, float isc) {
    float xn0 = (px - o0) * isc;
    float xn1 = (py - o1) * isc;
    float xn2 = (pz - o2) * isc * RATIO;
    xn0 = fminf(fmaxf(xn0, -1.0f), 1.0f);
    xn1 = fminf(fmaxf(xn1, -1.0f), 1.0f);
    xn2 = fminf(fmaxf(xn2, -1.0f), 1.0f);
    float fx = (xn0 + 1.0f) * 0.5f * (WW - 1);
    float fy = (xn1 + 1.0f) * 0.5f * (HH - 1);
    float fz = (xn2 + 1.0f) * 0.5f * (DD - 1);
    float x0f = floorf(fx), y0f = floorf(fy), z0f = floorf(fz);
    float tx = fx - x0f, ty = fy - y0f, tz = fz - z0f;
    int x0 = min(max((int)x0f,     0), WW - 1);
    int x1 = min(max((int)x0f + 1, 0), WW - 1);
    int y0 = min(max((int)y0f,     0), HH - 1);
    int y1 = min(max((int)y0f + 1, 0), HH - 1);
    int z0 = min(max((int)z0f,     0), DD - 1);
    int z1 = min(max((int)z0f + 1, 0), DD - 1);
    Tri t;
    int b00 = z0 * (HH * WW) + y0 * WW;
    int b01 = z0 * (HH * WW) + y1 * WW;
    int b10 = z1 * (HH * WW) + y0 * WW;
    int b11 = z1 * (HH * WW) + y1 * WW;
    t.idx[0] = b00 + x0; t.w[0] = (1-tz)*(1-ty)*(1-tx);
    t.idx[1] = b00 + x1; t.w[1] = (1-tz)*(1-ty)*tx;
    t.idx[2] = b01 + x0; t.w[2] = (1-tz)*ty*(1-tx);
    t.idx[3] = b01 + x1; t.w[3] = (1-tz)*ty*tx;
    t.idx[4] = b10 + x0; t.w[4] = tz*(1-ty)*(1-tx);
    t.idx[5] = b10 + x1; t.w[5] = tz*(1-ty)*tx;
    t.idx[6] = b11 + x0; t.w[6] = tz*ty*(1-tx);
    t.idx[7] = b11 + x1; t.w[7] = tz*ty*tx;
    return t;
}

__device__ inline void tri_gather(const float* __restrict__ vox, const Tri& t,
                                  int nch, float* out) {
    for (int c = 0; c < nch; ++c) {
        const float* p = vox + (size_t)c * DHW;
        float s = 0.0f;
        #pragma unroll
        for (int k = 0; k < 8; ++k) s += t.w[k] * p[t.idx[k]];
        out[c] = s;
    }
}

// ---------------- Broyden search: one thread per (point, init-bone) ----------------
__global__ void __launch_bounds__(256) k_search(
    const float* __restrict__ xd, const float* __restrict__ offset,
    const float* __restrict__ scale, const float* __restrict__ tinv,
    const float* __restrict__ vd, const float* __restrict__ vj,
    float* __restrict__ out_xc, float* __restrict__ out_J,
    float* __restrict__ raw_valid, int Npts)
{
    int m = blockIdx.x * blockDim.x + threadIdx.x;
    if (m >= Npts * NI) return;
    int n = m / NI;
    int i = m - n * NI;

    float p0 = xd[3*n], p1 = xd[3*n+1], p2 = xd[3*n+2];
    const float* tv = tinv + i * 12;
    float xc0 = tv[0]*p0 + tv[1]*p1 + tv[2]*p2  + tv[3];
    float xc1 = tv[4]*p0 + tv[5]*p1 + tv[6]*p2  + tv[7];
    float xc2 = tv[8]*p0 + tv[9]*p1 + tv[10]*p2 + tv[11];

    float o0 = offset[0], o1 = offset[1], o2 = offset[2];
    float isc = 1.0f / scale[0];

    Tri tr = tri_make(xc0, xc1, xc2, o0, o1, o2, isc);
    float g[3]; tri_gather(vd, tr, 3, g);
    g[0] -= p0; g[1] -= p1; g[2] -= p2;
    float Ji[9]; tri_gather(vj, tr, 9, Ji);

    bool cv = sqrtf(g[0]*g[0] + g[1]*g[1] + g[2]*g[2]) < CVG;
    bool dv = false;

    for (int it = 0; it < NITER; ++it) {
        if (!cv && !dv) {
            float dx0 = -(Ji[0]*g[0] + Ji[1]*g[1] + Ji[2]*g[2]);
            float dx1 = -(Ji[3]*g[0] + Ji[4]*g[1] + Ji[5]*g[2]);
            float dx2 = -(Ji[6]*g[0] + Ji[7]*g[1] + Ji[8]*g[2]);
            float nx0 = xc0 + dx0, nx1 = xc1 + dx1, nx2 = xc2 + dx2;
            Tri t2 = tri_make(nx0, nx1, nx2, o0, o1, o2, isc);
            float gn[3]; tri_gather(vd, t2, 3, gn);
            gn[0] -= p0; gn[1] -= p1; gn[2] -= p2;
            float dg0 = gn[0]-g[0], dg1 = gn[1]-g[1], dg2 = gn[2]-g[2];
            float Jd0 = Ji[0]*dg0 + Ji[1]*dg1 + Ji[2]*dg2;
            float Jd1 = Ji[3]*dg0 + Ji[4]*dg1 + Ji[5]*dg2;
            float Jd2 = Ji[6]*dg0 + Ji[7]*dg1 + Ji[8]*dg2;
            float denom = dx0*Jd0 + dx1*Jd1 + dx2*Jd2;
            if (fabsf(denom) < 1e-12f) denom = 1e-12f;
            float rd = 1.0f / denom;
            float u0 = (dx0 - Jd0) * rd, u1 = (dx1 - Jd1) * rd, u2 = (dx2 - Jd2) * rd;
            float vT0 = dx0*Ji[0] + dx1*Ji[3] + dx2*Ji[6];
            float vT1 = dx0*Ji[1] + dx1*Ji[4] + dx2*Ji[7];
            float vT2 = dx0*Ji[2] + dx1*Ji[5] + dx2*Ji[8];
            Ji[0]+=u0*vT0; Ji[1]+=u0*vT1; Ji[2]+=u0*vT2;
            Ji[3]+=u1*vT0; Ji[4]+=u1*vT1; Ji[5]+=u1*vT2;
            Ji[6]+=u2*vT0; Ji[7]+=u2*vT1; Ji[8]+=u2*vT2;
            xc0 = nx0; xc1 = nx1; xc2 = nx2;
            g[0] = gn[0]; g[1] = gn[1]; g[2] = gn[2];
            float nn = sqrtf(g[0]*g[0] + g[1]*g[1] + g[2]*g[2]);
            cv = nn < CVG;
            dv = nn > DVG;
        }
    }

    float xn0 = (xc0 - o0) * isc;
    float xn1 = (xc1 - o1) * isc;
    float xn2 = (xc2 - o2) * isc * RATIO;
    bool inside = (fabsf(xn0) <= 1.0f) && (fabsf(xn1) <= 1.0f) && (fabsf(xn2) <= 1.0f);
    bool okv = cv && !dv && inside;

    out_xc[3*m + 0] = xc0;
    out_xc[3*m + 1] = xc1;
    out_xc[3*m + 2] = xc2;
    #pragma unroll
    for (int k = 0; k < 9; ++k) out_J[(size_t)9*m + k] = Ji[k];
    raw_valid[m] = okv ? 1.0f : 0.0f;
}

// ---------------- dedup across init bones (reads stable raw-valid buffer) ----------------
__global__ void k_dedup(const float* __restrict__ xc, const float* __restrict__ raw,
                        float* __restrict__ out_valid, int Npts)
{
    int m = blockIdx.x * blockDim.x + threadIdx.x;
    if (m >= Npts * NI) return;
    int n = m / NI;
    int i = m - n * NI;
    float a0 = xc[3*m], a1 = xc[3*m+1], a2 = xc[3*m+2];
    bool dup = false;
    for (int j = 0; j < i; ++j) {
        int mj = n * NI + j;
        if (raw[mj] > 0.5f) {
            float d0 = a0 - xc[3*mj];
            float d1 = a1 - xc[3*mj+1];
            float d2 = a2 - xc[3*mj+2];
            if (d0*d0 + d1*d1 + d2*d2 < DUP_EPS2) dup = true;
        }
    }
    out_valid[m] = (raw[m] > 0.5f && !dup) ? 1.0f : 0.0f;
}

extern "C" void kernel_launch(void* const* d_in, const int* in_sizes, int n_in,
                              void* d_out, int out_size, void* d_ws, size_t ws_size,
                              hipStream_t stream) {
    const float* xd     = (const float*)d_in[0];
    const float* tfs    = (const float*)d_in[1];
    const float* lbs    = (const float*)d_in[2];
    const float* offset = (const float*)d_in[3];
    const float* scale  = (const float*)d_in[4];
    float* out = (float*)d_out;
    float* ws  = (float*)d_ws;

    const int Npts = in_sizes[0] / 3;        // 200000
    const int M    = Npts * NI;              // 2.6M

    // workspace layout (floats)
    float* ws_tinv = ws;                                // 13*12, padded to 256
    float* ws_vd   = ws + 256;                          // 3*DHW
    float* ws_vj   = ws + 256 + (size_t)3 * DHW;        // 9*DHW
    float* ws_raw  = ws + 256 + (size_t)12 * DHW;       // M

    // output layout: xc (M*3) | valid (M) | Jinv (M*9)
    float* out_xc    = out;
    float* out_valid = out + (size_t)3 * M;
    float* out_J     = out + (size_t)4 * M;

    k_tinv<<<1, 32, 0, stream>>>(tfs, ws_tinv);
    k_precompute<<<DHW / 128, 256, 0, stream>>>(lbs, tfs, offset, scale, ws_vd, ws_vj);
    k_search<<<(M + 255) / 256, 256, 0, stream>>>(xd, offset, scale, ws_tinv,
                                                  ws_vd, ws_vj, out_xc, out_J, ws_raw, Npts);
    k_dedup<<<(M + 255) / 256, 256, 0, stream>>>(out_xc, ws_raw, out_valid, Npts);
}